// _Mamba2ResidualBlock_22797686407840
// MI455X (gfx1250) — compile-verified
//
#include <hip/hip_runtime.h>
#include <hip/hip_bf16.h>
#include <math.h>

// ---------------- problem constants (from reference setup) ----------------
#define BATCH     128
#define D_MODEL   4096
#define D_INNER   8192
#define NH        128
#define DH        64
#define DS        128
#define NG        8
#define KCONV     4
#define CONV_DIM  (D_INNER + 2*NG*DS)            // 10240
#define PROJ_DIM  (2*D_INNER + 2*NG*DS + NH)     // 18560
#define EPSF      1e-5f

typedef float v2f __attribute__((ext_vector_type(2)));
typedef float v8f __attribute__((ext_vector_type(8)));

// =========================================================================
// WMMA fp32 GEMM:  C[M=128, N] = A[128, K] * W[N, K]^T  (+ optional R)
//
// Block = 256 threads = 8 waves; each wave owns one 16-wide N tile and all
// 8 M-tiles (full M=128), so the streamed weight W is read exactly once.
// A is staged cooperatively into double-buffered LDS (shared by all 8 waves,
// 8x fewer global A loads); W fragments are software-pipelined in registers.
// The inner WMMA loop touches only LDS + registers -> no global load waits
// between v_wmma issues.
//
// Fragment layouts per CDNA5 ISA 7.12.2:
//   A 16x4 : lane L (M = L&15) holds {A[m,k+2*half], A[m,k+2*half+1]}
//   B 4x16 : lane L (N = L&15) holds {W[n,k+2*half], W[n,k+2*half+1]}
//   C 16x16: vgpr r -> M = 8*half + r, N = L&15
// =========================================================================
#define KSTEP      32
#define LDS_STRIDE 36   // 128-row A chunk, padded: 16B-aligned float4 stores,
                        // and l16*36 mod 64 banks distinct for fragment reads

__global__ __launch_bounds__(256)
void wmma_gemm_f32_kernel(const float* __restrict__ A,   // 128 x K
                          const float* __restrict__ W,   // N x K
                          float* __restrict__ C,         // 128 x N
                          const float* __restrict__ R,   // 128 x N residual or null
                          int N, int K)
{
    __shared__ float sA[2][128 * LDS_STRIDE];            // 2 x 18KB ping-pong

    const int tid  = threadIdx.x;
    const int lane = tid & 31;
    const int wave = tid >> 5;
    const int n0   = (blockIdx.x * 8 + wave) * 16;       // grid tiles N exactly

    const int half = lane >> 4;     // 0: K0/1 + C rows M0..7 ; 1: K2/3 + M8..15
    const int l16  = lane & 15;
    const int koff = half * 2;

    v8f acc[8] = {};                // 8 M-tiles x (16x16 f32 / 32 lanes)

    // cooperative A-staging map: thread -> (row, 16-col half-chunk)
    const int arow = tid >> 1;
    const int acol = (tid & 1) * 16;
    const float* aptr  = A + (size_t)arow * (size_t)K + acol;
    float* sdst0 = &sA[0][arow * LDS_STRIDE + acol];
    float* sdst1 = &sA[1][arow * LDS_STRIDE + acol];

    const float* wrow = W + (size_t)(n0 + l16) * (size_t)K;  // this lane's W row

    // ---- prologue: stage chunk 0 (A -> LDS0, W -> regs) ----
    {
        float4 st0, st1, st2, st3;
        st0 = *(const float4*)(aptr + 0);
        st1 = *(const float4*)(aptr + 4);
        st2 = *(const float4*)(aptr + 8);
        st3 = *(const float4*)(aptr + 12);
        *(float4*)(sdst0 + 0)  = st0;
        *(float4*)(sdst0 + 4)  = st1;
        *(float4*)(sdst0 + 8)  = st2;
        *(float4*)(sdst0 + 12) = st3;
    }
    v2f wreg[8];
#pragma unroll
    for (int j = 0; j < 8; ++j)
        wreg[j] = *(const v2f*)(wrow + j * 4 + koff);
    __syncthreads();

    const int nchunks = K / KSTEP;
    for (int c = 0; c < nchunks; ++c) {
        const int kbase  = c * KSTEP;
        const float* sbuf = sA[c & 1];
        const bool more   = (c + 1 < nchunks);

        // ---- issue next chunk's global loads (overlap with WMMA below) ----
        float4 st0, st1, st2, st3;
        v2f wnext[8];
        if (more) {
            const float* ap = aptr + kbase + KSTEP;
            __builtin_prefetch(wrow + kbase + 2 * KSTEP, 0, 3);
            st0 = *(const float4*)(ap + 0);
            st1 = *(const float4*)(ap + 4);
            st2 = *(const float4*)(ap + 8);
            st3 = *(const float4*)(ap + 12);
#pragma unroll
            for (int j = 0; j < 8; ++j)
                wnext[j] = *(const v2f*)(wrow + kbase + KSTEP + j * 4 + koff);
        }

        // ---- compute: 8 k-substeps x 8 M-tiles, A from LDS, W from regs ----
#pragma unroll
        for (int kk = 0; kk < 8; ++kk) {
            const v2f bfrag = wreg[kk];
#pragma unroll
            for (int t = 0; t < 8; ++t) {
                const float* sa = sbuf + (t * 16 + l16) * LDS_STRIDE + kk * 4 + koff;
                v2f afrag = *(const v2f*)sa;
                acc[t] = __builtin_amdgcn_wmma_f32_16x16x4_f32(
                    /*neg_a=*/false, afrag,
                    /*neg_b=*/false, bfrag,
                    /*c_mod=*/(short)0, acc[t],
                    /*reuse_a=*/false, /*reuse_b=*/false);
            }
        }

        __syncthreads();   // all waves done reading this chunk's LDS buffer
        if (more) {
            float* sdst = (c & 1) ? sdst0 : sdst1;
            *(float4*)(sdst + 0)  = st0;
            *(float4*)(sdst + 4)  = st1;
            *(float4*)(sdst + 8)  = st2;
            *(float4*)(sdst + 12) = st3;
#pragma unroll
            for (int j = 0; j < 8; ++j) wreg[j] = wnext[j];
            __syncthreads();
        }
    }

    // ---- epilogue: store C (optionally fused residual add) ----
    const int n = n0 + l16;
#pragma unroll
    for (int t = 0; t < 8; ++t) {
#pragma unroll
        for (int r = 0; r < 8; ++r) {
            const int m = t * 16 + half * 8 + r;
            float v = acc[t][r];
            if (R) v += R[(size_t)m * N + n];
            C[(size_t)m * N + n] = v;
        }
    }
}

// =========================================================================
// RMSNorm: one block per batch row
// =========================================================================
__global__ __launch_bounds__(256)
void rmsnorm_kernel(const float* __restrict__ x, const float* __restrict__ w,
                    float* __restrict__ out, int n)
{
    const int b = blockIdx.x;
    const float* row = x + (size_t)b * n;
    float ss = 0.f;
    for (int i = threadIdx.x; i < n; i += blockDim.x) { float v = row[i]; ss += v * v; }
    __shared__ float red[256];
    red[threadIdx.x] = ss; __syncthreads();
    for (int s = 128; s > 0; s >>= 1) {
        if ((int)threadIdx.x < s) red[threadIdx.x] += red[threadIdx.x + s];
        __syncthreads();
    }
    const float scale = rsqrtf(red[0] / (float)n + EPSF);
    for (int i = threadIdx.x; i < n; i += blockDim.x)
        out[(size_t)b * n + i] = row[i] * scale * w[i];
}

// =========================================================================
// Depthwise conv (K=4) + SiLU + new_conv_state shift-in
// =========================================================================
__global__ __launch_bounds__(256)
void conv_silu_kernel(const float* __restrict__ proj,
                      const float* __restrict__ conv_state,
                      const float* __restrict__ conv_w,
                      const float* __restrict__ conv_b,
                      float* __restrict__ new_conv_state,
                      float* __restrict__ xbc_out)
{
    const int idx = blockIdx.x * blockDim.x + threadIdx.x;  // b*CONV_DIM + c
    const int b = idx / CONV_DIM;
    const int c = idx % CONV_DIM;

    const float4 w = *(const float4*)(conv_w + (size_t)c * KCONV);
    const float* cs = conv_state + ((size_t)b * CONV_DIM + c) * (KCONV - 1);
    const float s0 = cs[0], s1 = cs[1], s2 = cs[2];
    const float xin = proj[(size_t)b * PROJ_DIM + D_INNER + c];

    float acc = s0 * w.x + s1 * w.y + s2 * w.z + xin * w.w + conv_b[c];
    const float sil = acc / (1.f + __expf(-acc));
    xbc_out[(size_t)b * CONV_DIM + c] = sil;

    float* ncs = new_conv_state + ((size_t)b * CONV_DIM + c) * (KCONV - 1);
    ncs[0] = s1; ncs[1] = s2; ncs[2] = xin;
}

// =========================================================================
// dt = softplus(dt + bias); dA = exp(dt * -exp(A_log))
// =========================================================================
__global__ __launch_bounds__(256)
void dtda_kernel(const float* __restrict__ proj,
                 const float* __restrict__ dt_bias,
                 const float* __restrict__ A_log,
                 float* __restrict__ dt_out, float* __restrict__ dA_out)
{
    const int idx = blockIdx.x * blockDim.x + threadIdx.x;  // b*NH + h
    const int b = idx / NH;
    const int h = idx % NH;
    const float raw = proj[(size_t)b * PROJ_DIM + D_INNER + CONV_DIM + h] + dt_bias[h];
    const float dt  = (raw > 20.f) ? raw : log1pf(__expf(raw));
    dt_out[idx] = dt;
    dA_out[idx] = __expf(dt * -__expf(A_log[h]));
}

// =========================================================================
// SSM state update + y = einsum('bhds,bhs->bhd') -- the 1 GB streaming kernel.
// One block per (b,h): 64x128 fp32 state tile, float4 (b128) loads/stores,
// B/C/x staged in LDS, 4-lane shuffle reduction over the s dimension.
// =========================================================================
__global__ __launch_bounds__(256)
void ssm_update_kernel(const float* __restrict__ ssm_state,
                       const float* __restrict__ xbc,
                       const float* __restrict__ dt_buf,
                       const float* __restrict__ dA_buf,
                       const float* __restrict__ D_param,
                       float* __restrict__ new_state,
                       float* __restrict__ y_out)
{
    const int bh = blockIdx.x;           // b*NH + h
    const int b  = bh / NH;
    const int h  = bh % NH;
    const int g  = h / (NH / NG);

    __shared__ float sB[DS], sC[DS], sx[DH];
    const float* base = xbc + (size_t)b * CONV_DIM;
    if (threadIdx.x < DS) {
        sB[threadIdx.x] = base[D_INNER + g * DS + threadIdx.x];
        sC[threadIdx.x] = base[D_INNER + NG * DS + g * DS + threadIdx.x];
    }
    if (threadIdx.x < DH) sx[threadIdx.x] = base[(size_t)h * DH + threadIdx.x];
    __syncthreads();

    const float dtv = dt_buf[bh];
    const float dAv = dA_buf[bh];

    const int d   = threadIdx.x >> 2;         // 0..63
    const int sb  = (threadIdx.x & 3) * 32;   // s segment
    const float xd  = sx[d];
    const float dtx = dtv * xd;

    const size_t off = ((size_t)bh * DH + d) * DS + sb;
    const float4* sp = (const float4*)(ssm_state + off);
    float4*       np = (float4*)(new_state + off);

    float ysum = 0.f;
#pragma unroll
    for (int i = 0; i < 8; ++i) {
        const float4 sv = sp[i];
        const int s = sb + i * 4;
        float4 nv;
        nv.x = sv.x * dAv + dtx * sB[s + 0];
        nv.y = sv.y * dAv + dtx * sB[s + 1];
        nv.z = sv.z * dAv + dtx * sB[s + 2];
        nv.w = sv.w * dAv + dtx * sB[s + 3];
        ysum += nv.x * sC[s + 0] + nv.y * sC[s + 1] + nv.z * sC[s + 2] + nv.w * sC[s + 3];
        np[i] = nv;
    }
    ysum += __shfl_xor(ysum, 1, 32);
    ysum += __shfl_xor(ysum, 2, 32);
    if ((threadIdx.x & 3) == 0)
        y_out[(size_t)b * D_INNER + h * DH + d] = ysum + D_param[h] * xd;
}

// =========================================================================
// y *= silu(z); then RMSNorm(y) * gate_norm_w.  One block per batch row.
// =========================================================================
__global__ __launch_bounds__(256)
void gate_norm_kernel(const float* __restrict__ y,
                      const float* __restrict__ proj,
                      const float* __restrict__ gw,
                      float* __restrict__ out)
{
    const int b = blockIdx.x;
    const float* yr = y    + (size_t)b * D_INNER;
    const float* zr = proj + (size_t)b * PROJ_DIM;   // z = proj[:, :d_inner]
    float* orow = out + (size_t)b * D_INNER;

    float ss = 0.f;
    for (int i = threadIdx.x; i < D_INNER; i += blockDim.x) {
        const float z = zr[i];
        const float v = yr[i] * (z / (1.f + __expf(-z)));
        orow[i] = v;
        ss += v * v;
    }
    __shared__ float red[256];
    red[threadIdx.x] = ss; __syncthreads();
    for (int s = 128; s > 0; s >>= 1) {
        if ((int)threadIdx.x < s) red[threadIdx.x] += red[threadIdx.x + s];
        __syncthreads();
    }
    const float scale = rsqrtf(red[0] / (float)D_INNER + EPSF);
    for (int i = threadIdx.x; i < D_INNER; i += blockDim.x)
        orow[i] = orow[i] * scale * gw[i];
}

// =========================================================================
// launch
// =========================================================================
extern "C" void kernel_launch(void* const* d_in, const int* in_sizes, int n_in,
                              void* d_out, int out_size, void* d_ws, size_t ws_size,
                              hipStream_t stream) {
    const float* hidden_states = (const float*)d_in[0];   // 128 x 1 x 4096
    const float* conv_state    = (const float*)d_in[1];   // 128 x 10240 x 3
    const float* ssm_state     = (const float*)d_in[2];   // 128 x 128 x 64 x 128
    const float* norm_w        = (const float*)d_in[3];   // 4096
    const float* in_proj_w     = (const float*)d_in[4];   // 18560 x 4096
    const float* conv_w        = (const float*)d_in[5];   // 10240 x 4
    const float* conv_b        = (const float*)d_in[6];   // 10240
    const float* dt_bias       = (const float*)d_in[7];   // 128
    const float* A_log         = (const float*)d_in[8];   // 128
    const float* D_param       = (const float*)d_in[9];   // 128
    const float* gate_norm_w   = (const float*)d_in[10];  // 8192
    const float* out_proj_w    = (const float*)d_in[11];  // 4096 x 8192

    // outputs, concatenated flat in return order
    float* out_hidden = (float*)d_out;                                        // 128*4096
    float* out_conv   = out_hidden + (size_t)BATCH * D_MODEL;                 // 128*10240*3
    float* out_ssm    = out_conv + (size_t)BATCH * CONV_DIM * (KCONV - 1);    // 128*128*64*128

    // workspace layout (floats)
    float* ws      = (float*)d_ws;
    float* normed  = ws;                                        // 128*4096
    float* proj    = normed  + (size_t)BATCH * D_MODEL;         // 128*18560
    float* xbc     = proj    + (size_t)BATCH * PROJ_DIM;        // 128*10240
    float* dt_buf  = xbc     + (size_t)BATCH * CONV_DIM;        // 128*128
    float* dA_buf  = dt_buf  + (size_t)BATCH * NH;              // 128*128
    float* y_buf   = dA_buf  + (size_t)BATCH * NH;              // 128*8192
    float* y_norm  = y_buf   + (size_t)BATCH * D_INNER;         // 128*8192

    // 1) input RMSNorm
    rmsnorm_kernel<<<BATCH, 256, 0, stream>>>(hidden_states, norm_w, normed, D_MODEL);

    // 2) in_proj: (128x4096)*(4096x18560) via V_WMMA_F32_16X16X4_F32
    //    18560/16 = 1160 N-tiles, 8 tiles per 256-thread block -> 145 blocks
    wmma_gemm_f32_kernel<<<PROJ_DIM / (16 * 8), 256, 0, stream>>>(
        normed, in_proj_w, proj, nullptr, PROJ_DIM, D_MODEL);

    // 3) depthwise conv + SiLU + conv-state shift
    conv_silu_kernel<<<(BATCH * CONV_DIM) / 256, 256, 0, stream>>>(
        proj, conv_state, conv_w, conv_b, out_conv, xbc);

    // 4) dt softplus / dA exp
    dtda_kernel<<<(BATCH * NH) / 256, 256, 0, stream>>>(proj, dt_bias, A_log, dt_buf, dA_buf);

    // 5) SSM state update + contraction (dominant 1 GB stream)
    ssm_update_kernel<<<BATCH * NH, 256, 0, stream>>>(
        ssm_state, xbc, dt_buf, dA_buf, D_param, out_ssm, y_buf);

    // 6) gate + gated RMSNorm
    gate_norm_kernel<<<BATCH, 256, 0, stream>>>(y_buf, proj, gate_norm_w, y_norm);

    // 7) out_proj: (128x8192)*(8192x4096) + residual, fused in WMMA epilogue
    wmma_gemm_f32_kernel<<<D_MODEL / (16 * 8), 256, 0, stream>>>(
        y_norm, out_proj_w, out_hidden, hidden_states, D_MODEL, D_INNER);
}